// DBI_44985487458968
// MI455X (gfx1250) — compile-verified
//
#include <hip/hip_runtime.h>

// Davies-Bouldin index, single data pass via sufficient statistics.
//   ws layout (floats):
//     [0      .. 6399]  S1   : per-cluster coordinate sums   [100][64]
//     [6400   .. 6499]  S2   : per-cluster sum of ||x||^2    [100]
//     [6500   .. 6599]  CNT  : per-cluster raw point count   [100]
//     [6600   .. 13767] AI   : centroids, zero-padded to 112 rows [112][64]
//     [13768  .. 13879] SI   : per-cluster scatter, zero-padded   [112]
#define K_CLUS 100
#define D_FEAT 64
#define K_PAD  112                         // 7 * 16 WMMA tiles
#define SUM_F  (K_CLUS * D_FEAT)           // 6400
#define WS_S2  SUM_F                       // 6400
#define WS_CNT (SUM_F + K_CLUS)            // 6500
#define WS_AI  (SUM_F + 2 * K_CLUS)        // 6600
#define WS_SI  (WS_AI + K_PAD * D_FEAT)    // 13768
#define WS_TOTAL (WS_SI + K_PAD)           // 13880 floats

typedef __attribute__((ext_vector_type(2))) float v2f;
typedef __attribute__((ext_vector_type(8))) float v8f;

// ---------------------------------------------------------------------------
// Kernel 1: one pass over 512 MB. Wave-per-point; lane l loads float2 = dims
// {2l, 2l+1} (one coalesced 256B row per wave-iteration). LDS-privatized
// scatter-add (ds_add_f32, conflict-free: LDS word index mod 64 == dim), then
// one global-atomic flush of 6600 floats per block.
// ---------------------------------------------------------------------------
__global__ void dbi_accum(const float* __restrict__ x,
                          const int* __restrict__ clus,
                          float* __restrict__ ws, int n) {
    __shared__ float l[SUM_F + 2 * K_CLUS];
    const int tid = threadIdx.x;
    for (int i = tid; i < SUM_F + 2 * K_CLUS; i += blockDim.x) l[i] = 0.f;
    __syncthreads();

    const int lane = tid & 31;
    const int wavesPerBlk = blockDim.x >> 5;
    const int wave = blockIdx.x * wavesPerBlk + (tid >> 5);
    const int nwaves = gridDim.x * wavesPerBlk;

    for (int p = wave; p < n; p += nwaves) {
        const int c = clus[p];                       // 0..99
        float2 v = ((const float2*)(x + (size_t)p * D_FEAT))[lane];
        atomicAdd(&l[c * D_FEAT + 2 * lane],     v.x);
        atomicAdd(&l[c * D_FEAT + 2 * lane + 1], v.y);
        float s = v.x * v.x + v.y * v.y;             // wave-reduce ||x||^2
        for (int off = 16; off > 0; off >>= 1) s += __shfl_xor(s, off, 32);
        if (lane == 0) {
            atomicAdd(&l[WS_S2 + c], s);
            atomicAdd(&l[WS_CNT + c], 1.f);
        }
    }
    __syncthreads();
    for (int i = tid; i < SUM_F + 2 * K_CLUS; i += blockDim.x)
        atomicAdd(&ws[i], l[i]);
}

// ---------------------------------------------------------------------------
// Kernel 2 (tiny): centroids + per-cluster scatter Si from sufficient stats.
//   Ai = (0.001 + S1) / (1 + cnt)
//   sum_sq = S2 - 2*Ai.S1 + cnt*||Ai||^2 ;  Si = sqrt((0.001+sum_sq)/(1+cnt))
// ---------------------------------------------------------------------------
__global__ void dbi_finalize(float* __restrict__ ws) {
    __shared__ float Al[SUM_F];
    const int tid = threadIdx.x;
    for (int i = tid; i < SUM_F; i += blockDim.x) {
        const int k = i >> 6;
        const float a = (0.001f + ws[i]) / (1.f + ws[WS_CNT + k]);
        Al[i] = a;
        ws[WS_AI + i] = a;       // rows 100..111 stay zero from the memset
    }
    __syncthreads();
    for (int k = tid; k < K_CLUS; k += blockDim.x) {
        float dot = 0.f, nr = 0.f;
        for (int d = 0; d < D_FEAT; ++d) {
            const float a = Al[k * D_FEAT + d];
            dot += a * ws[k * D_FEAT + d];
            nr  += a * a;
        }
        const float cnt = ws[WS_CNT + k];
        const float ssq = fmaxf(0.001f + ws[WS_S2 + k] - 2.f * dot + cnt * nr, 0.f);
        ws[WS_SI + k] = sqrtf(ssq / (1.f + cnt));
    }
}

// ---------------------------------------------------------------------------
// Kernel 3: Gram matrix G = Ai*Ai^T via V_WMMA_F32_16X16X4_F32 (fp32 WMMA, no
// precision loss). 112x112 -> 7x7 tiles of 16x16, K-loop of 16 WMMA steps.
// One workgroup, 8 waves, ~6 tiles/wave. Then Mij^2 = nrm_i+nrm_j-2G_ij,
// Rij = (Si+Sj)/Mij (masked), row-max via LDS int atomicMax (all values >= 0),
// block-reduce -> DBI scalar.
//
// Fragment layouts (ISA 7.12.2, 32-bit):
//   A 16x4 : lane<16 -> (M=lane, K=k0+{0,1}) ; lane>=16 -> (M=lane-16, K=k0+{2,3})
//   B 4x16 : N-striped mirror; for B = A^T this is the same indexing with the
//            column tile's row base.
//   C 16x16: vgpr r, lane<16 -> (M=r, N=lane) ; lane>=16 -> (M=r+8, N=lane-16)
// ---------------------------------------------------------------------------
__global__ void dbi_pairwise(const float* __restrict__ ws, float* __restrict__ out) {
    __shared__ float As[K_PAD * D_FEAT];   // 28.7 KB centroids (zero-padded)
    __shared__ float nrm[K_PAD];
    __shared__ float si[K_PAD];
    __shared__ int   rmax[K_PAD];
    __shared__ float part[8];

    const int tid = threadIdx.x;
    for (int i = tid; i < K_PAD * D_FEAT; i += blockDim.x) As[i] = ws[WS_AI + i];
    for (int i = tid; i < K_PAD; i += blockDim.x) { si[i] = ws[WS_SI + i]; rmax[i] = 0; }
    __syncthreads();
    for (int k = tid; k < K_PAD; k += blockDim.x) {
        float s = 0.f;
        for (int d = 0; d < D_FEAT; ++d) { const float a = As[k * D_FEAT + d]; s += a * a; }
        nrm[k] = s;
    }
    __syncthreads();

    const int lane = tid & 31;
    const int wave = tid >> 5;
    const int row  = lane & 15;
    const int koff = (lane >> 4) << 1;             // 0 for lanes 0-15, 2 for 16-31

    for (int t = wave; t < 49; t += 8) {           // wave-uniform loop: EXEC all-1s
        const int i0 = (t / 7) * 16;
        const int j0 = (t % 7) * 16;
        v8f c = {0.f, 0.f, 0.f, 0.f, 0.f, 0.f, 0.f, 0.f};
        #pragma unroll
        for (int kk = 0; kk < D_FEAT; kk += 4) {
            v2f a, b;
            a[0] = As[(i0 + row) * D_FEAT + kk + koff];
            a[1] = As[(i0 + row) * D_FEAT + kk + koff + 1];
            b[0] = As[(j0 + row) * D_FEAT + kk + koff];
            b[1] = As[(j0 + row) * D_FEAT + kk + koff + 1];
            c = __builtin_amdgcn_wmma_f32_16x16x4_f32(
                    false, a, false, b, (short)0, c, false, false);
        }
        const int iBase = i0 + ((lane < 16) ? 0 : 8);
        const int j = j0 + row;
        #pragma unroll
        for (int r = 0; r < 8; ++r) {
            const int i = iBase + r;
            if (i < K_CLUS && j < K_CLUS && i != j) {
                const float d2  = fmaxf(nrm[i] + nrm[j] - 2.f * c[r], 0.f);
                const float rij = (si[i] + si[j]) / sqrtf(d2);
                atomicMax(&rmax[i], __float_as_int(rij));   // rij >= 0
            }
        }
    }
    __syncthreads();

    float v = (tid < K_CLUS) ? __int_as_float(rmax[tid]) : 0.f;
    for (int off = 16; off > 0; off >>= 1) v += __shfl_xor(v, off, 32);
    if (lane == 0) part[wave] = v;
    __syncthreads();
    if (tid == 0) {
        float s = 0.f;
        for (int w = 0; w < 8; ++w) s += part[w];
        out[0] = s / (float)K_CLUS;
    }
}

extern "C" void kernel_launch(void* const* d_in, const int* in_sizes, int n_in,
                              void* d_out, int out_size, void* d_ws, size_t ws_size,
                              hipStream_t stream) {
    const float* x    = (const float*)d_in[0];
    const int*   clus = (const int*)d_in[1];   // JAX default: int64 -> int32
    float*       out  = (float*)d_out;
    float*       ws   = (float*)d_ws;
    const int n = in_sizes[0] / D_FEAT;

    hipMemsetAsync(d_ws, 0, WS_TOTAL * sizeof(float), stream);  // capture-safe
    dbi_accum   <<<1024, 256, 0, stream>>>(x, clus, ws, n);
    dbi_finalize<<<1,    256, 0, stream>>>(ws);
    dbi_pairwise<<<1,    256, 0, stream>>>(ws, out);
}